// _LightGCN_16879221473502
// MI455X (gfx1250) — compile-verified
//
#include <hip/hip_runtime.h>

typedef __attribute__((ext_vector_type(2))) float v2f;
typedef __attribute__((ext_vector_type(8))) float v8f;

#define EMBED_DIM 64

// ---------------------------------------------------------------------------
// ego = concat(user_emb, item_emb); acc = ego; cur = ego; nxt = 0
// float4-vectorized over the whole node table.
// ---------------------------------------------------------------------------
__global__ void lgcn_init(const float* __restrict__ user_emb,
                          const float* __restrict__ item_emb,
                          float* __restrict__ acc,
                          float* __restrict__ cur,
                          float* __restrict__ nxt,
                          int num_users, size_t total4) {
  size_t i = (size_t)blockIdx.x * blockDim.x + threadIdx.x;  // float4 index
  if (i >= total4) return;
  size_t node = i / (EMBED_DIM / 4);
  size_t q    = i % (EMBED_DIM / 4);
  float4 v;
  if (node < (size_t)num_users)
    v = ((const float4*)user_emb)[node * (EMBED_DIM / 4) + q];
  else
    v = ((const float4*)item_emb)[(node - num_users) * (EMBED_DIM / 4) + q];
  ((float4*)acc)[i] = v;
  ((float4*)cur)[i] = v;
  ((float4*)nxt)[i] = make_float4(0.f, 0.f, 0.f, 0.f);
}

// ---------------------------------------------------------------------------
// SpMM scatter: next[rows[e], :] += vals[e] * cur[cols[e], :]
// 32 lanes per edge, float2 gather (L2-resident: table is 38.4MB < 192MB L2),
// two f32 global atomics per lane.
// ---------------------------------------------------------------------------
__global__ void lgcn_spmm(const int* __restrict__ rows,
                          const int* __restrict__ cols,
                          const float* __restrict__ vals,
                          const float* __restrict__ cur,
                          float* __restrict__ nxt,
                          long long nnz) {
  long long idx = (long long)blockIdx.x * blockDim.x + threadIdx.x;
  long long total = nnz * 32;
  if (idx >= total) return;
  long long e = idx >> 5;
  int h = (int)(idx & 31);                  // which float2 of the 64-dim row
  int r = rows[e];
  int c = cols[e];
  float v = vals[e];
  const float2 x = *(const float2*)(cur + (size_t)c * EMBED_DIM + h * 2);
  float* dst = nxt + (size_t)r * EMBED_DIM + h * 2;
  atomicAdd(dst,     v * x.x);
  atomicAdd(dst + 1, v * x.y);
}

// ---------------------------------------------------------------------------
// acc += src, and (optionally) zero the buffer that will receive the next
// layer's scatter. float4-vectorized.
// ---------------------------------------------------------------------------
__global__ void lgcn_accum(float* __restrict__ acc,
                           const float* __restrict__ src,
                           float* __restrict__ tozero,
                           size_t total4) {
  size_t i = (size_t)blockIdx.x * blockDim.x + threadIdx.x;
  if (i >= total4) return;
  float4 a = ((float4*)acc)[i];
  float4 s = ((const float4*)src)[i];
  a.x += s.x; a.y += s.y; a.z += s.z; a.w += s.w;
  ((float4*)acc)[i] = a;
  if (tozero) ((float4*)tozero)[i] = make_float4(0.f, 0.f, 0.f, 0.f);
}

// ---------------------------------------------------------------------------
// Final scoring: out[b] = dot(acc[users[b]], acc[NU+items[b]]) / 16
// One wave32 per group of 16 pairs. D = U(16x4) x I^T(4x16) accumulated over
// K=64 via V_WMMA_F32_16X16X4_F32 (full f32 precision); diagonal = dots.
//
// A layout (16x4 f32, ISA 7.12.2): lane L<16 holds row M=L, {K0,K1};
// lanes 16-31 hold row M=L-16, {K2,K3}. B (4x16) mirrors with N=lane&15.
// ---------------------------------------------------------------------------
__global__ void lgcn_dot_wmma(const float* __restrict__ acc,
                              const int* __restrict__ users,
                              const int* __restrict__ items,
                              float* __restrict__ out,
                              int num_users, int ngroups) {
  int wave = (int)((blockIdx.x * blockDim.x + threadIdx.x) >> 5);
  int lane = threadIdx.x & 31;
  if (wave >= ngroups) return;              // wave-uniform: EXEC stays all-1s

  int r    = lane & 15;                     // row (A) / col (B) index
  int half = lane >> 4;                     // 0 -> K{0,1}, 1 -> K{2,3}

  int u  = users[wave * 16 + r];
  int it = items[wave * 16 + r];
  const float* urow = acc + (size_t)u * EMBED_DIM;
  const float* irow = acc + (size_t)(num_users + it) * EMBED_DIM;

  v8f c = {0.f, 0.f, 0.f, 0.f, 0.f, 0.f, 0.f, 0.f};
#pragma unroll
  for (int kb = 0; kb < EMBED_DIM; kb += 4) {
    v2f a = *(const v2f*)(urow + kb + half * 2);
    v2f b = *(const v2f*)(irow + kb + half * 2);
    // (neg_a, A, neg_b, B, c_mod, C, reuse_a, reuse_b)
    c = __builtin_amdgcn_wmma_f32_16x16x4_f32(false, a, false, b,
                                              (short)0, c, false, false);
  }

  // Diagonal extraction (C/D layout: VGPR v, lanes 0-15 -> M=v, N=lane;
  // lanes 16-31 -> M=v+8, N=lane-16).
  const float scale = 1.0f / 16.0f;         // (acc/4)·(acc/4)
  int m, vidx; bool active;
  if (lane < 8)        { m = lane;      vidx = lane;      active = true;  }
  else if (lane >= 24) { m = lane - 16; vidx = lane - 24; active = true;  }
  else                 { m = 0;         vidx = 0;         active = false; }
  float d = c[0];
#pragma unroll
  for (int v = 1; v < 8; ++v) d = (vidx == v) ? c[v] : d;
  if (active) out[wave * 16 + m] = d * scale;
}

// ---------------------------------------------------------------------------
// Host launcher
// ---------------------------------------------------------------------------
extern "C" void kernel_launch(void* const* d_in, const int* in_sizes, int n_in,
                              void* d_out, int out_size, void* d_ws, size_t ws_size,
                              hipStream_t stream) {
  const int*   users    = (const int*)d_in[0];
  const int*   items    = (const int*)d_in[1];
  const int*   adj_rows = (const int*)d_in[2];
  const int*   adj_cols = (const int*)d_in[3];
  const float* adj_vals = (const float*)d_in[4];
  const float* user_emb = (const float*)d_in[5];
  const float* item_emb = (const float*)d_in[6];

  int num_users = in_sizes[5] / EMBED_DIM;
  int num_items = in_sizes[6] / EMBED_DIM;
  int n_nodes   = num_users + num_items;
  long long nnz = (long long)in_sizes[2];
  int batch     = out_size;

  size_t node_f = (size_t)n_nodes * EMBED_DIM;   // floats per table
  float* acc = (float*)d_ws;
  float* c0  = acc + node_f;
  float* c1  = c0 + node_f;

  const int tb = 256;
  size_t total4 = node_f / 4;
  dim3 g_tab((unsigned)((total4 + tb - 1) / tb));
  dim3 g_spmm((unsigned)((nnz * 32 + tb - 1) / tb));

  lgcn_init<<<g_tab, tb, 0, stream>>>(user_emb, item_emb, acc, c0, c1,
                                      num_users, total4);

  // layer 1: c0 -> c1 (c1 zeroed by init); then acc += c1, zero c0
  lgcn_spmm<<<g_spmm, tb, 0, stream>>>(adj_rows, adj_cols, adj_vals, c0, c1, nnz);
  lgcn_accum<<<g_tab, tb, 0, stream>>>(acc, c1, c0, total4);

  // layer 2: c1 -> c0; then acc += c0, zero c1
  lgcn_spmm<<<g_spmm, tb, 0, stream>>>(adj_rows, adj_cols, adj_vals, c1, c0, nnz);
  lgcn_accum<<<g_tab, tb, 0, stream>>>(acc, c0, c1, total4);

  // layer 3: c0 -> c1; then acc += c1 (no more zeroing needed)
  lgcn_spmm<<<g_spmm, tb, 0, stream>>>(adj_rows, adj_cols, adj_vals, c0, c1, nnz);
  lgcn_accum<<<g_tab, tb, 0, stream>>>(acc, c1, nullptr, total4);

  // Final WMMA scoring: one wave per 16 pairs
  int ngroups = batch / 16;
  dim3 g_dot((unsigned)((ngroups * 32 + tb - 1) / tb));
  lgcn_dot_wmma<<<g_dot, tb, 0, stream>>>(acc, users, items, (float*)d_out,
                                          num_users, ngroups);
}